// DeStationaryCausalAttention_42159398977542
// MI455X (gfx1250) — compile-verified
//
#include <hip/hip_runtime.h>
#include <hip/hip_bf16.h>

#define DD   1024
#define HH   16
#define HDIM 64
#define DKV  512
#define KVHD 32
#define HID  128
#define BB   2
#define LL   2048
#define FF   32

typedef __attribute__((ext_vector_type(2))) float v2f;
typedef __attribute__((ext_vector_type(8))) float v8f;
typedef __attribute__((ext_vector_type(4))) unsigned int u32x4;
typedef __attribute__((ext_vector_type(4))) int i32x4;
typedef __attribute__((ext_vector_type(8))) int i32x8;

// ---------------------------------------------------------------------------
// TDM: issue a 2D tile load (global -> LDS) via TENSOR_LOAD_TO_LDS.
// D# layout per CDNA5 ISA ch.8 (group0 128b, group1 256b; groups 2/3 unused
// for 2D). data_size = 4 bytes. Optional LDS row padding (bank-conflict
// avoidance) done by the TDM itself via pad_interval/pad_amount.
// ---------------------------------------------------------------------------
__device__ __forceinline__ void tdm_load_2d_f32(
    unsigned lds_off_bytes, const void* gptr,
    unsigned tensor_w, unsigned tensor_h, unsigned row_stride,
    unsigned tile_w, unsigned tile_h,
    unsigned pad_en, unsigned pi, unsigned pa)
{
  const unsigned long long ga = (unsigned long long)(uintptr_t)gptr;
  u32x4 g0;
  g0[0] = 1u;                                   // count=1 (valid), user mode
  g0[1] = lds_off_bytes;                        // lds_addr
  g0[2] = (unsigned)ga;                         // global_addr[31:0]
  g0[3] = (unsigned)((ga >> 32) & 0x1FFFFFFull) // global_addr[56:32]
          | 0x80000000u;                        // type=2 ("image") in bits 127:126
  i32x8 g1;
  g1[0] = (int)((2u << 16)                      // data_size = 4B
          | (pad_en << 20) | (pi << 22) | (pa << 25));
  g1[1] = (int)((tensor_w & 0xFFFFu) << 16);                          // dim0[15:0]
  g1[2] = (int)((tensor_w >> 16) | ((tensor_h & 0xFFFFu) << 16));     // dim0[31:16] | dim1[15:0]
  g1[3] = (int)((tensor_h >> 16) | (tile_w << 16));                   // dim1[31:16] | tile_dim0
  g1[4] = (int)tile_h;                                                // tile_dim1 (tile_dim2=0)
  g1[5] = (int)row_stride;                                            // dim0_stride[31:0]
  g1[6] = 0;                                                          // dim0_stride[47:32] | dim1_stride lo
  g1[7] = 0;
  i32x4 z4 = {0, 0, 0, 0};
#if defined(__clang_major__) && __clang_major__ >= 23
  i32x8 z8 = {0, 0, 0, 0, 0, 0, 0, 0};
  __builtin_amdgcn_tensor_load_to_lds(g0, g1, z4, z4, z8, 0);
#else
  __builtin_amdgcn_tensor_load_to_lds(g0, g1, z4, z4, 0);
#endif
}

// ---------------------------------------------------------------------------
// Kernel 1: C[M,N] = A[M,K] * W[K,N]  (row-major f32), V_WMMA_F32_16X16X4_F32.
// Block = 128 threads (4 waves), block tile 64(M) x 64(N), K-chunk 16.
// TDM double-buffers the A (64x16, LDS stride 17) and B (16x64, LDS stride 65)
// tiles; wave 0 drives the DMA, all waves compute from LDS.
// Dynamic LDS layout (floats):  A0[0..1087] A1[1088..2175] B0[2176..3215] B1[3216..4255]
// ---------------------------------------------------------------------------
#define KC     16
#define A_STR  17
#define B_STR  65
#define A_OFF1 1088
#define B_OFF0 2176
#define B_STEP 1040          // B_OFF1 - B_OFF0
#define GEMM_LDS_FLOATS 4256

__global__ __launch_bounds__(128) void wmma_gemm_f32(
    const float* __restrict__ A, const float* __restrict__ W,
    float* __restrict__ C, int M, int Kd, int N)
{
  extern __shared__ float smem[];   // no static LDS -> dynamic base offset 0

  const int lane = threadIdx.x & 31;
  const int wave = threadIdx.x >> 5;
  const int hi   = lane >> 4;       // 0: K0/K1 half, 1: K2/K3 half
  const int r    = lane & 15;

  const int n_base = blockIdx.y * 64;
  const int m_base = blockIdx.x * 64;
  const int n0     = n_base + wave * 16;

  const int niter = Kd / KC;

  // Prologue: wave 0 kicks off the first A/B tiles through the TDM.
  if (wave == 0) {
    tdm_load_2d_f32(0u, A + (size_t)m_base * Kd,
                    (unsigned)Kd, (unsigned)M, (unsigned)Kd,
                    KC, 64, 1u, 3u, 0u);     // pad 1 DW per 16 DW row -> stride 17
    tdm_load_2d_f32(B_OFF0 * 4u, W + n_base,
                    (unsigned)N, (unsigned)Kd, (unsigned)N,
                    64, KC, 1u, 5u, 0u);     // pad 1 DW per 64 DW row -> stride 65
  }

  v8f c0 = {}, c1 = {}, c2 = {}, c3 = {};

  for (int it = 0; it < niter; ++it) {
    const int cur = it & 1, nxt = cur ^ 1;

    __builtin_amdgcn_s_wait_tensorcnt(0);   // wave 0: tiles for `cur` landed
    __syncthreads();                        // publish LDS; prev reads of `nxt` done

    if (wave == 0 && (it + 1) < niter) {
      const int k1 = (it + 1) * KC;
      tdm_load_2d_f32((unsigned)(nxt * A_OFF1) * 4u,
                      A + (size_t)m_base * Kd + k1,
                      (unsigned)Kd, (unsigned)M, (unsigned)Kd,
                      KC, 64, 1u, 3u, 0u);
      tdm_load_2d_f32((unsigned)(B_OFF0 + nxt * B_STEP) * 4u,
                      W + (size_t)k1 * N + n_base,
                      (unsigned)N, (unsigned)Kd, (unsigned)N,
                      64, KC, 1u, 5u, 0u);
    }

    const float* At = smem + cur * A_OFF1;            // integer offset select
    const float* Bt = smem + B_OFF0 + cur * B_STEP;   // (no pointer aggregates)
    #pragma unroll
    for (int kk = 0; kk < KC; kk += 4) {
      const int ka = kk + 2 * hi;
      v2f bf;
      bf[0] = Bt[ka * B_STR + wave * 16 + r];
      bf[1] = Bt[(ka + 1) * B_STR + wave * 16 + r];
      v2f a0; a0[0] = At[(r     ) * A_STR + ka]; a0[1] = At[(r     ) * A_STR + ka + 1];
      v2f a1; a1[0] = At[(r + 16) * A_STR + ka]; a1[1] = At[(r + 16) * A_STR + ka + 1];
      v2f a2; a2[0] = At[(r + 32) * A_STR + ka]; a2[1] = At[(r + 32) * A_STR + ka + 1];
      v2f a3; a3[0] = At[(r + 48) * A_STR + ka]; a3[1] = At[(r + 48) * A_STR + ka + 1];

      c0 = __builtin_amdgcn_wmma_f32_16x16x4_f32(false, a0, false, bf, (short)0, c0, false, false);
      c1 = __builtin_amdgcn_wmma_f32_16x16x4_f32(false, a1, false, bf, (short)0, c1, false, false);
      c2 = __builtin_amdgcn_wmma_f32_16x16x4_f32(false, a2, false, bf, (short)0, c2, false, false);
      c3 = __builtin_amdgcn_wmma_f32_16x16x4_f32(false, a3, false, bf, (short)0, c3, false, false);
    }
  }

  const int col = n0 + r;
  #pragma unroll
  for (int i = 0; i < 8; ++i) {
    const int row = m_base + i + hi * 8;
    C[(size_t)(row     ) * N + col] = c0[i];
    C[(size_t)(row + 16) * N + col] = c1[i];
    C[(size_t)(row + 32) * N + col] = c2[i];
    C[(size_t)(row + 48) * N + col] = c3[i];
  }
}

// ---------------------------------------------------------------------------
// Kernel 2: per-batch "last position" prep: tau scalar, delta_last (512),
// qc = first 32 cols of each 64-wide head slice of h_last @ Wq.
// ---------------------------------------------------------------------------
__device__ __forceinline__ float gelu_exact(float x) {
  return 0.5f * x * (1.0f + erff(x * 0.70710678118654752f));
}

__global__ __launch_bounds__(128) void prep_kernel(
    const float* __restrict__ h, const float* __restrict__ mu,
    const float* __restrict__ sigma, const float* __restrict__ Wq,
    const float* __restrict__ tw1, const float* __restrict__ tb1,
    const float* __restrict__ tw2, const float* __restrict__ tb2,
    const float* __restrict__ dw1, const float* __restrict__ db1,
    const float* __restrict__ dw2, const float* __restrict__ db2,
    float* __restrict__ qc, float* __restrict__ tauo, float* __restrict__ dlast)
{
  __shared__ float hsh[DD];
  __shared__ float tg[HID], dg[HID];
  __shared__ float s_sig, s_mu;

  const int b   = blockIdx.x;
  const int tid = threadIdx.x;
  const size_t last = (size_t)b * LL + (LL - 1);
  const float* hl = h + last * DD;

  for (int i = tid; i < DD; i += blockDim.x) hsh[i] = hl[i];
  if (tid == 0) {
    float ss = 0.f, sm = 0.f;
    for (int i = 0; i < FF; ++i) {
      ss += sigma[last * FF + i];
      sm += mu[last * FF + i];
    }
    s_sig = fmaxf(ss / FF, 1e-6f);
    s_mu  = sm / FF;
  }
  __syncthreads();

  for (int j = tid; j < HID; j += blockDim.x) {
    float at = tb1[j] + s_sig * tw1[j];   // row 0 = concatenated mean feature
    float ad = db1[j] + s_mu  * dw1[j];
    for (int i = 0; i < DD; ++i) {
      const float x = hsh[i];
      at += x * tw1[(size_t)(i + 1) * HID + j];
      ad += x * dw1[(size_t)(i + 1) * HID + j];
    }
    tg[j] = gelu_exact(at);
    dg[j] = gelu_exact(ad);
  }
  __syncthreads();

  if (tid == 0) {
    float t = tb2[0];
    for (int j = 0; j < HID; ++j) t += tg[j] * tw2[j];
    t = fminf(fmaxf(t, -3.f), 3.f);
    tauo[b] = expf(t);
  }
  for (int d = tid; d < DKV; d += blockDim.x) {
    float v = db2[d];
    for (int j = 0; j < HID; ++j) v += dg[j] * dw2[(size_t)j * DKV + d];
    dlast[(size_t)b * DKV + d] = fminf(fmaxf(v, -5.f), 5.f);
  }
  for (int c = tid; c < HH * KVHD; c += blockDim.x) {
    const int head = c >> 5, dd = c & 31;
    const int col = head * HDIM + dd;   // only first 32 of each 64-wide head
    float q = 0.f;
    for (int i = 0; i < DD; ++i) q += hsh[i] * Wq[(size_t)i * DD + col];
    qc[(size_t)b * HH * KVHD + c] = q;
  }
}

// ---------------------------------------------------------------------------
// Kernel 3: last-query attention per (b, head). Fused logit weight:
//   w = tau * qc / sqrt(32) + delta_last ;  logit_k = clip(w . K_k, +-50)
// Softmax over 2048 keys, then ctx = sum_k p_k * V_k (32 dims).
// ---------------------------------------------------------------------------
__global__ __launch_bounds__(256) void attn_last_kernel(
    const float* __restrict__ Kws, const float* __restrict__ Vws,
    const float* __restrict__ qc, const float* __restrict__ tau,
    const float* __restrict__ dlast, float* __restrict__ ctx)
{
  __shared__ float ls[LL];
  __shared__ float red[8];
  __shared__ float sacc[KVHD];
  __shared__ float s_max, s_sum;

  const int bh  = blockIdx.x;
  const int b   = bh / HH, hh = bh % HH;
  const int tid = threadIdx.x;
  const int lane = tid & 31, wv = tid >> 5;

  if (tid < KVHD) sacc[tid] = 0.f;

  const float t   = tau[b];
  const float inv = 0.17677669529663689f;   // 1/sqrt(32)
  float w[KVHD];
  #pragma unroll
  for (int d = 0; d < KVHD; ++d)
    w[d] = t * qc[(size_t)b * HH * KVHD + hh * KVHD + d] * inv +
           dlast[(size_t)b * DKV + hh * KVHD + d];

  const float* Kb = Kws + (size_t)b * LL * DKV + hh * KVHD;
  const float* Vb = Vws + (size_t)b * LL * DKV + hh * KVHD;

  float mx = -1e30f;
  for (int k = tid; k < LL; k += 256) {
    const float* kr = Kb + (size_t)k * DKV;
    float s = 0.f;
    #pragma unroll
    for (int d = 0; d < KVHD; ++d) s += w[d] * kr[d];
    s = fminf(fmaxf(s, -50.f), 50.f);
    ls[k] = s;
    mx = fmaxf(mx, s);
  }
  #pragma unroll
  for (int off = 16; off; off >>= 1) mx = fmaxf(mx, __shfl_xor(mx, off, 32));
  if (lane == 0) red[wv] = mx;
  __syncthreads();
  if (tid == 0) {
    float m = red[0];
    for (int i = 1; i < 8; ++i) m = fmaxf(m, red[i]);
    s_max = m;
  }
  __syncthreads();

  const float m = s_max;
  float sum = 0.f;
  for (int k = tid; k < LL; k += 256) {
    const float e = expf(ls[k] - m);
    ls[k] = e;
    sum += e;
  }
  #pragma unroll
  for (int off = 16; off; off >>= 1) sum += __shfl_xor(sum, off, 32);
  if (lane == 0) red[wv] = sum;
  __syncthreads();
  if (tid == 0) {
    float s = 0.f;
    for (int i = 0; i < 8; ++i) s += red[i];
    s_sum = s;
  }
  __syncthreads();

  float acc[KVHD];
  #pragma unroll
  for (int d = 0; d < KVHD; ++d) acc[d] = 0.f;
  for (int k = tid; k < LL; k += 256) {
    const float p = ls[k];
    const float* vr = Vb + (size_t)k * DKV;
    #pragma unroll
    for (int d = 0; d < KVHD; ++d) acc[d] += p * vr[d];
  }
  #pragma unroll
  for (int d = 0; d < KVHD; ++d) {
    #pragma unroll
    for (int off = 16; off; off >>= 1) acc[d] += __shfl_xor(acc[d], off, 32);
  }
  if (lane == 0) {
    #pragma unroll
    for (int d = 0; d < KVHD; ++d) atomicAdd(&sacc[d], acc[d]);
  }
  __syncthreads();
  if (tid < KVHD)
    ctx[(size_t)b * DKV + hh * KVHD + tid] = sacc[tid] / s_sum;
}

// ---------------------------------------------------------------------------
// Kernel 4: out[b, n] = ctx[b, :] @ Wo[:, n] + bo[n]
// ---------------------------------------------------------------------------
__global__ __launch_bounds__(256) void out_proj_kernel(
    const float* __restrict__ ctx, const float* __restrict__ Wo,
    const float* __restrict__ bo, float* __restrict__ out)
{
  const int idx = blockIdx.x * blockDim.x + threadIdx.x;
  if (idx >= BB * DD) return;
  const int b = idx / DD, n = idx % DD;
  const float* c = ctx + (size_t)b * DKV;
  float s = bo[n];
  for (int d = 0; d < DKV; ++d) s += c[d] * Wo[(size_t)d * DD + n];
  out[idx] = s;
}

extern "C" void kernel_launch(void* const* d_in, const int* in_sizes, int n_in,
                              void* d_out, int out_size, void* d_ws, size_t ws_size,
                              hipStream_t stream) {
  const float* h    = (const float*)d_in[0];
  const float* mu   = (const float*)d_in[1];
  const float* sig  = (const float*)d_in[2];
  const float* Wq   = (const float*)d_in[3];
  const float* Wk   = (const float*)d_in[4];
  const float* Wv   = (const float*)d_in[5];
  const float* Wo   = (const float*)d_in[6];
  const float* bo   = (const float*)d_in[7];
  const float* tw1  = (const float*)d_in[8];
  const float* tb1  = (const float*)d_in[9];
  const float* tw2  = (const float*)d_in[10];
  const float* tb2  = (const float*)d_in[11];
  const float* dw1  = (const float*)d_in[12];
  const float* db1  = (const float*)d_in[13];
  const float* dw2  = (const float*)d_in[14];
  const float* db2  = (const float*)d_in[15];

  float* ws   = (float*)d_ws;
  float* Kws  = ws;                                   // B*L*DKV
  float* Vws  = Kws + (size_t)BB * LL * DKV;          // B*L*DKV
  float* qc   = Vws + (size_t)BB * LL * DKV;          // B*H*32
  float* tauw = qc + BB * HH * KVHD;                  // B (+pad)
  float* dl   = tauw + 16;                            // B*DKV
  float* ctx  = dl + BB * DKV;                        // B*DKV

  const int M = BB * LL;  // 4096
  dim3 ggrid(M / 64, DKV / 64);  // (64, 8)
  const size_t lds_bytes = GEMM_LDS_FLOATS * sizeof(float);  // 17024 B
  wmma_gemm_f32<<<ggrid, 128, lds_bytes, stream>>>(h, Wk, Kws, M, DD, DKV);
  wmma_gemm_f32<<<ggrid, 128, lds_bytes, stream>>>(h, Wv, Vws, M, DD, DKV);

  prep_kernel<<<BB, 128, 0, stream>>>(h, mu, sig, Wq,
                                      tw1, tb1, tw2, tb2,
                                      dw1, db1, dw2, db2,
                                      qc, tauw, dl);

  attn_last_kernel<<<BB * HH, 256, 0, stream>>>(Kws, Vws, qc, tauw, dl, ctx);

  out_proj_kernel<<<(BB * DD + 255) / 256, 256, 0, stream>>>(ctx, Wo, bo, (float*)d_out);
}